// PointLoss_55688545960492
// MI455X (gfx1250) — compile-verified
//
#include <hip/hip_runtime.h>
#include <hip/hip_bf16.h>
#include <stdint.h>

typedef __attribute__((ext_vector_type(2))) float v2f;
typedef __attribute__((ext_vector_type(8))) float v8f;

#define N_PTS 4096
#define TILE  128          // block tile (rows and cols)
#define BLK   256          // threads per block (8 wave32)
#define BIGF  1e30f
#define MASK_OR 0x7F000000u   // OR'd into bits of masked-out candidates (> bits(1e30))

__device__ __forceinline__ unsigned int umin2(unsigned int a, unsigned int b) {
    return a < b ? a : b;   // v_min_u32
}

// ---- workspace layout (bytes) ----
// half1   : float[4096*4]   @ 0        (h = reflected points, xyzw, w=0)
// ptsNeg2 : float[4096*4]   @ 65536    (-2 * points, xyzw, w=0)
// normA   : float[4096]     @ 131072   (|h|^2)
// normB   : float[4096]     @ 147456   (|p|^2)
// m1      : uint [4096]     @ 163840
// d1bits  : uint [4096]     @ 180224   (column mins, bit-pattern)
// d2bits  : uint [4096]     @ 196608   (row mins, bit-pattern)

// ---------------- Phase 1: per-point prep ----------------
__global__ void pl_prep(const float* __restrict__ norm,
                        const float* __restrict__ points,
                        float* __restrict__ half1,
                        float* __restrict__ ptsNeg2,
                        float* __restrict__ normA,
                        float* __restrict__ normB,
                        unsigned int* __restrict__ m1,
                        unsigned int* __restrict__ d1bits,
                        unsigned int* __restrict__ d2bits)
{
    int i = blockIdx.x * blockDim.x + threadIdx.x;
    if (i >= N_PTS) return;

    float nx = norm[0], ny = norm[1], nz = norm[2], nd = norm[3];
    float s  = nx*nx + ny*ny + nz*nz;

    float px = points[3*i+0], py = points[3*i+1], pz = points[3*i+2];
    float two_side = -2.0f * (px*nx + py*ny + pz*nz + nd);
    float k  = two_side / s;
    float hx = px + k*nx, hy = py + k*ny, hz = pz + k*nz;

    half1[4*i+0] = hx; half1[4*i+1] = hy; half1[4*i+2] = hz; half1[4*i+3] = 0.0f;
    ptsNeg2[4*i+0] = -2.0f*px; ptsNeg2[4*i+1] = -2.0f*py;
    ptsNeg2[4*i+2] = -2.0f*pz; ptsNeg2[4*i+3] = 0.0f;
    normA[i] = hx*hx + hy*hy + hz*hz;
    normB[i] = px*px + py*py + pz*pz;
    m1[i]    = (two_side > 0.0f) ? 1u : 0u;

    unsigned int big = __float_as_uint(BIGF);
    d1bits[i] = big;
    d2bits[i] = big;
}

// ---------------- Phase 2: WMMA distance tiles + hierarchical min ----------------
__global__ __launch_bounds__(BLK)
void pl_tiles(const float* __restrict__ half1,
              const float* __restrict__ ptsNeg2,
              const float* __restrict__ normA,
              const float* __restrict__ normB,
              const unsigned int* __restrict__ m1,
              unsigned int* __restrict__ d1bits,
              unsigned int* __restrict__ d2bits)
{
    __shared__ float sA[TILE*4];     // h rows (xyzw)
    __shared__ float sB[TILE*4];     // -2p cols (xyzw)
    __shared__ float sNA[TILE];
    __shared__ float sNB[TILE];
    __shared__ unsigned int sOrA[TILE];  // 0 if row in m1 else MASK_OR
    __shared__ unsigned int sOrB[TILE];  // 0 if col in m2 else MASK_OR
    __shared__ unsigned int sD1[TILE];   // per-column min (this block)
    __shared__ unsigned int sD2[TILE];   // per-row min (this block)

    const int j0 = blockIdx.x * TILE;   // columns
    const int i0 = blockIdx.y * TILE;   // rows
    const int t  = threadIdx.x;
    const unsigned int big = __float_as_uint(BIGF);

    if (t < TILE) {
        *(float4*)&sA[t*4] = *(const float4*)&half1[(i0 + t)*4];
        sNA[t]  = normA[i0 + t];
        sOrA[t] = m1[i0 + t] ? 0u : MASK_OR;   // rows must be in m1
        sD1[t]  = big;
    } else {
        int tt = t - TILE;
        *(float4*)&sB[tt*4] = *(const float4*)&ptsNeg2[(j0 + tt)*4];
        sNB[tt]  = normB[j0 + tt];
        sOrB[tt] = m1[j0 + tt] ? MASK_OR : 0u; // cols must be in m2 = !m1
        sD2[tt]  = big;
    }
    __syncthreads();

    const int wave = t >> 5;          // 0..7 (wave32)
    const int lane = t & 31;
    const int lr   = lane & 15;       // row/col within fragment
    const int kk   = (lane >> 4) * 2; // K pair selector per ISA A/B 32-bit layout

    // A fragment: rows [16*wave, 16*wave+16), K = kk,kk+1
    const int rowA = 16*wave + lr;
    v2f afrag;
    afrag.x = sA[rowA*4 + kk + 0];
    afrag.y = sA[rowA*4 + kk + 1];

    // Hoist row-side (M) LDS data into registers: invariant across c-loop.
    // C/D layout: VGPR r -> M = r + (lane>=16 ? 8 : 0); N = lane&15
    const int mBase = 16*wave + ((lane >> 4) << 3);
    float na[8];
    unsigned int orA[8];
    unsigned int rowMin[8];
    #pragma unroll
    for (int r = 0; r < 8; ++r) {
        na[r]     = sNA[mBase + r];
        orA[r]    = sOrA[mBase + r];
        rowMin[r] = big;
    }

    #pragma unroll
    for (int c = 0; c < 8; ++c) {
        // B fragment: B[K][N] = (-2p)[col][K]
        const int colB = 16*c + lr;
        v2f bfrag;
        bfrag.x = sB[colB*4 + kk + 0];
        bfrag.y = sB[colB*4 + kk + 1];

        const int nLoc = 16*c + lr;
        const float nb = sNB[nLoc];
        const unsigned int orB = sOrB[nLoc];

        // C[m][n] = |h_m|^2 + |p_n|^2 so the WMMA emits D = ||h_m - p_n||^2 directly.
        v8f cfrag;
        #pragma unroll
        for (int r = 0; r < 8; ++r) cfrag[r] = na[r] + nb;

        v8f acc = __builtin_amdgcn_wmma_f32_16x16x4_f32(
                      false, afrag, false, bfrag, (short)0, cfrag, false, false);

        // Integer-domain epilogue: for x >= 0, uint(bits) ordering == float ordering.
        unsigned int colMin = big;
        #pragma unroll
        for (int r = 0; r < 8; ++r) {
            int bi = __float_as_int(acc[r]);
            unsigned int bits = (unsigned int)(bi < 0 ? 0 : bi); // v_max_i32: clamp <0 to +0
            colMin    = umin2(colMin,    bits | orA[r]);         // masked col candidate
            rowMin[r] = umin2(rowMin[r], bits | orB);            // masked row candidate
        }
        atomicMin(&sD1[nLoc], colMin);                 // 1 ds_min_u32 per c
    }

    #pragma unroll
    for (int r = 0; r < 8; ++r)
        atomicMin(&sD2[mBase + r], rowMin[r]);         // 8 ds_min_u32 total
    __syncthreads();

    if (t < TILE) {
        atomicMin(&d1bits[j0 + t], sD1[t]);            // global_atomic_min_u32
    } else {
        atomicMin(&d2bits[i0 + (t - TILE)], sD2[t - TILE]);
    }
}

// ---------------- Phase 3: final masked means -> scalar ----------------
__global__ __launch_bounds__(BLK)
void pl_final(const unsigned int* __restrict__ d1bits,
              const unsigned int* __restrict__ d2bits,
              const unsigned int* __restrict__ m1,
              float* __restrict__ out)
{
    __shared__ float rs1[BLK], rs2[BLK];
    __shared__ unsigned int rc1[BLK], rc2[BLK];

    const int t = threadIdx.x;
    float s1 = 0.0f, s2 = 0.0f;
    unsigned int c1 = 0, c2 = 0;

    for (int i = t; i < N_PTS; i += BLK) {
        unsigned int m1v = m1[i];
        unsigned int m2v = 1u - m1v;
        float d1v = __uint_as_float(d1bits[i]);
        float d2v = __uint_as_float(d2bits[i]);
        if (m2v) { s1 += d1v; c2++; }
        if (m1v) { s2 += d2v; c1++; }
    }
    rs1[t] = s1; rs2[t] = s2; rc1[t] = c1; rc2[t] = c2;
    __syncthreads();

    for (int off = BLK/2; off > 0; off >>= 1) {
        if (t < off) {
            rs1[t] += rs1[t+off];
            rs2[t] += rs2[t+off];
            rc1[t] += rc1[t+off];
            rc2[t] += rc2[t+off];
        }
        __syncthreads();
    }

    if (t == 0) {
        float fc1 = fmaxf((float)rc1[0], 1.0f);
        float fc2 = fmaxf((float)rc2[0], 1.0f);
        float av1 = rs1[0] / fc2;
        float av2 = rs2[0] / fc1;
        out[0] = (0.5f*av1 + 0.5f*av2) * 100.0f;
    }
}

extern "C" void kernel_launch(void* const* d_in, const int* in_sizes, int n_in,
                              void* d_out, int out_size, void* d_ws, size_t ws_size,
                              hipStream_t stream)
{
    const float* norm   = (const float*)d_in[0];   // (1,4) f32
    const float* points = (const float*)d_in[1];   // (4096,3) f32
    float* out = (float*)d_out;

    char* ws = (char*)d_ws;
    float* half1   = (float*)(ws + 0);
    float* ptsNeg2 = (float*)(ws + 65536);
    float* normA   = (float*)(ws + 131072);
    float* normB   = (float*)(ws + 147456);
    unsigned int* m1     = (unsigned int*)(ws + 163840);
    unsigned int* d1bits = (unsigned int*)(ws + 180224);
    unsigned int* d2bits = (unsigned int*)(ws + 196608);

    pl_prep<<<N_PTS / BLK, BLK, 0, stream>>>(norm, points, half1, ptsNeg2,
                                             normA, normB, m1, d1bits, d2bits);

    dim3 grid(N_PTS / TILE, N_PTS / TILE, 1);   // 32 x 32 blocks
    pl_tiles<<<grid, BLK, 0, stream>>>(half1, ptsNeg2, normA, normB,
                                       m1, d1bits, d2bits);

    pl_final<<<1, BLK, 0, stream>>>(d1bits, d2bits, m1, out);
}